// TimeAwareMultiHeadAttention_5351529250957
// MI455X (gfx1250) — compile-verified
//
#include <hip/hip_runtime.h>
#include <hip/hip_bf16.h>

// Time-aware MHA: B=16, L=256, H=128, NHEADS=4, d=32.
// Strategy:
//  1) proj_wmma: Y = X @ W^T + bias (+ addend), via V_WMMA_F32_16X16X4_F32.
//     Produces Qp, Ke = K+posK, Ve = V+posV in workspace (2 MB each).
//  2) attn_kernel: one block per (b,q). Streams tK/tV rows (the 2x537MB
//     tensors that dominate) with coalesced loads, causal-truncated, with
//     per-head wave32 reductions + LDS softmax. Pure-bandwidth kernel.

#define L 256
#define HID 128
#define NH 4
#define DH 32

typedef __attribute__((ext_vector_type(2))) float v2f;
typedef __attribute__((ext_vector_type(8))) float v8f;

// ---------------------------------------------------------------------------
// Kernel 1: tiled GEMM  Y[M,128] = X[M,128] @ W[128,128]^T + bias (+ addend)
// One wave computes one 16x16 output tile; 4 waves/block cover 4 N-tiles.
// grid = (M/16, 128/64), block = 128.
// ---------------------------------------------------------------------------
__global__ __launch_bounds__(128) void proj_wmma(const float* __restrict__ X,
                                                 const float* __restrict__ W,
                                                 const float* __restrict__ bias,
                                                 const float* __restrict__ addend,
                                                 float* __restrict__ Y) {
  const int lane = threadIdx.x & 31;
  const int wave = threadIdx.x >> 5;
  const int m0 = blockIdx.x * 16;
  const int n0 = (blockIdx.y * 4 + wave) * 16;

  // A fragment mapping (16x4 f32): lanes 0-15 -> M rows, VGPR {0,1} = K {0,1};
  // lanes 16-31 -> same M rows, K {2,3}.
  const int arow = m0 + (lane & 15);
  const int kb   = (lane >> 4) * 2;
  // B fragment mapping (4x16 f32): lanes 0-15 -> N cols with K {0,1},
  // lanes 16-31 -> same N cols with K {2,3}.  B[k][n] = W[n][k] (W row-major [out,in]).
  const int bcol = n0 + (lane & 15);

  v8f c = {};
  #pragma unroll 4
  for (int k0 = 0; k0 < HID; k0 += 4) {
    v2f a, b;
    a.x = X[arow * HID + k0 + kb];
    a.y = X[arow * HID + k0 + kb + 1];
    b.x = W[bcol * HID + k0 + kb];
    b.y = W[bcol * HID + k0 + kb + 1];
    // (neg_a, A, neg_b, B, c_mod, C, reuse_a, reuse_b)
    c = __builtin_amdgcn_wmma_f32_16x16x4_f32(false, a, false, b, (short)0, c,
                                              false, false);
  }

  // D layout: VGPR r holds M = m0 + r + 8*(lane>=16), N = n0 + (lane&15).
  const int rbase = m0 + 8 * (lane >> 4);
  const float bv = bias[bcol];
  #pragma unroll
  for (int r = 0; r < 8; ++r) {
    const int m = rbase + r;
    float v = c[r] + bv;
    if (addend) v += addend[m * HID + bcol];
    Y[m * HID + bcol] = v;
  }
}

// ---------------------------------------------------------------------------
// Kernel 2: attention. grid = (L, B), block = 128 (thread == feature index).
// wave w == head h (32 lanes == 32 d-elements of that head).
// ---------------------------------------------------------------------------
__global__ __launch_bounds__(128) void attn_kernel(
    const float* __restrict__ Qp, const float* __restrict__ Ke,
    const float* __restrict__ Ve, const unsigned char* __restrict__ time_mask,
    const float* __restrict__ tK, const float* __restrict__ tV,
    float* __restrict__ out) {
  const int q = blockIdx.x;
  const int b = blockIdx.y;
  const int tid = threadIdx.x;   // feature index 0..127
  const int lane = tid & 31;
  const int h = tid >> 5;

  __shared__ float w_lds[NH][L];  // per-head softmax weights

  const long bq = (long)b * L + q;
  const float* __restrict__ tKrow = tK + bq * (long)(L * HID);
  const float* __restrict__ tVrow = tV + bq * (long)(L * HID);
  const float* __restrict__ KeB = Ke + (long)b * L * HID;
  const float* __restrict__ VeB = Ve + (long)b * L * HID;

  // Pre-scale Q by 1/sqrt(d): all three logit terms multiply Q.
  const float qs = Qp[bq * HID + tid] * 0.17677669529663687f;  // 1/sqrt(32)
  const bool masked = (time_mask[bq] != 0);  // block-uniform

  float acc = 0.0f;

  if (!masked) {
    // ---- Pass A: logits[h][k] for k <= q (causal) ----
    #pragma unroll 4
    for (int k = 0; k <= q; ++k) {
      __builtin_prefetch(tKrow + (k + 16) * HID + tid, 0, 0);  // global_prefetch_b8
      float x = qs * (KeB[k * HID + tid] + tKrow[k * HID + tid]);
      // wave32 tree reduction over the head's 32 d-lanes
      #pragma unroll
      for (int off = 16; off; off >>= 1) x += __shfl_xor(x, off, 32);
      if (lane == 0) w_lds[h][k] = x;
    }
    __syncthreads();

    // ---- softmax over k in [0, q], one wave per head ----
    float mx = -3.0e38f;
    for (int k = lane; k <= q; k += 32) mx = fmaxf(mx, w_lds[h][k]);
    #pragma unroll
    for (int off = 16; off; off >>= 1) mx = fmaxf(mx, __shfl_xor(mx, off, 32));
    float sum = 0.0f;
    for (int k = lane; k <= q; k += 32) {
      float e = expf(w_lds[h][k] - mx);
      w_lds[h][k] = e;
      sum += e;
    }
    #pragma unroll
    for (int off = 16; off; off >>= 1) sum += __shfl_xor(sum, off, 32);
    const float inv = 1.0f / sum;
    for (int k = lane; k <= q; k += 32) w_lds[h][k] *= inv;
    __syncthreads();

    // ---- Pass B: out = sum_k w[k] * (Ve[k] + tV[k]) ----
    #pragma unroll 4
    for (int k = 0; k <= q; ++k) {
      __builtin_prefetch(tVrow + (k + 16) * HID + tid, 0, 0);
      acc = fmaf(w_lds[h][k], VeB[k * HID + tid] + tVrow[k * HID + tid], acc);
    }
  } else {
    // Fully-masked query row: softmax of constant NEG == uniform 1/L over ALL k.
    #pragma unroll 4
    for (int k = 0; k < L; ++k) {
      __builtin_prefetch(tVrow + (k + 16) * HID + tid, 0, 0);
      acc += VeB[k * HID + tid] + tVrow[k * HID + tid];
    }
    acc *= (1.0f / (float)L);
  }

  out[bq * HID + tid] = acc;
}

// ---------------------------------------------------------------------------
// Launch
// ---------------------------------------------------------------------------
extern "C" void kernel_launch(void* const* d_in, const int* in_sizes, int n_in,
                              void* d_out, int out_size, void* d_ws,
                              size_t ws_size, hipStream_t stream) {
  const float* queries = (const float*)d_in[0];            // [B,L,H]
  const float* keys    = (const float*)d_in[1];            // [B,L,H]
  const unsigned char* time_mask = (const unsigned char*)d_in[2];  // [B,L] bool
  // d_in[3] attn_mask: deterministic causal mask, recomputed in-kernel.
  const float* tK = (const float*)d_in[4];                 // [B,L,L,H]
  const float* tV = (const float*)d_in[5];                 // [B,L,L,H]
  const float* pK = (const float*)d_in[6];                 // [B,L,H]
  const float* pV = (const float*)d_in[7];                 // [B,L,H]
  const float* Qw = (const float*)d_in[8];
  const float* Qb = (const float*)d_in[9];
  const float* Kw = (const float*)d_in[10];
  const float* Kb = (const float*)d_in[11];
  const float* Vw = (const float*)d_in[12];
  const float* Vb = (const float*)d_in[13];
  float* out = (float*)d_out;

  const int B = 16;
  const int M = B * L;  // 4096 rows
  float* Qp = (float*)d_ws;        // [M,128]
  float* KeW = Qp + (size_t)M * HID;
  float* VeW = KeW + (size_t)M * HID;

  dim3 pgrid(M / 16, HID / 64);  // (256, 2)
  proj_wmma<<<pgrid, 128, 0, stream>>>(queries, Qw, Qb, nullptr, Qp);
  proj_wmma<<<pgrid, 128, 0, stream>>>(keys, Kw, Kb, pK, KeW);  // Ke = K + pK
  proj_wmma<<<pgrid, 128, 0, stream>>>(keys, Vw, Vb, pV, VeW);  // Ve = V + pV

  dim3 agrid(L, B);  // one block per (b,q)
  attn_kernel<<<agrid, 128, 0, stream>>>(Qp, KeW, VeW, time_mask, tK, tV, out);
}